// QuadraticNonLinearity_79139067396770
// MI455X (gfx1250) — compile-verified
//
#include <hip/hip_runtime.h>
#include <hip/hip_bf16.h>
#include <math.h>

typedef float v2f __attribute__((ext_vector_type(2)));
typedef float v8f __attribute__((ext_vector_type(8)));

// ---------------- CG setup (device, double precision, exact e3nn convention) ----------------

struct cplx { double re, im; };
__device__ inline cplx cmul(cplx a, cplx b) {
  cplx r; r.re = a.re * b.re - a.im * b.im; r.im = a.re * b.im + a.im * b.re; return r;
}
__device__ inline double dfact(int n) { double r = 1.0; for (int i = 2; i <= n; ++i) r *= (double)i; return r; }
__device__ inline int imax3(int a, int b, int c) { int m = a > b ? a : b; return m > c ? m : c; }
__device__ inline int imin3(int a, int b, int c) { int m = a < b ? a : b; return m < c ? m : c; }

__device__ double su2_cg(int j1, int m1, int j2, int m2, int j3, int m3) {
  if (m3 != m1 + m2) return 0.0;
  double pre = sqrt((2.0 * j3 + 1.0) * dfact(j3 + j1 - j2) * dfact(j3 - j1 + j2) *
                    dfact(j1 + j2 - j3) / dfact(j1 + j2 + j3 + 1));
  pre *= sqrt(dfact(j3 + m3) * dfact(j3 - m3) * dfact(j1 - m1) * dfact(j1 + m1) *
              dfact(j2 - m2) * dfact(j2 + m2));
  int kmin = imax3(0, j2 - j3 - m1, j1 - j3 + m2);
  int kmax = imin3(j1 + j2 - j3, j1 - m1, j2 + m2);
  double s = 0.0;
  for (int k = kmin; k <= kmax; ++k) {
    double d = dfact(k) * dfact(j1 + j2 - j3 - k) * dfact(j1 - m1 - k) *
               dfact(j2 + m2 - k) * dfact(j3 - j2 + m1 + k) * dfact(j3 - j1 - m2 + k);
    s += ((k & 1) ? -1.0 : 1.0) / d;
  }
  return pre * s;
}

__device__ void buildQ(int l, cplx* q) {
  int d = 2 * l + 1;
  for (int i = 0; i < d * d; ++i) { q[i].re = 0.0; q[i].im = 0.0; }
  double rs = 1.0 / sqrt(2.0);
  for (int m = -l; m < 0; ++m) {
    q[(l + m) * d + (l - m)].re = rs;      // q[l+m, l+|m|] = 1/sqrt2
    q[(l + m) * d + (l + m)].im = -rs;     // q[l+m, l-|m|] = -i/sqrt2
  }
  q[l * d + l].re = 1.0;
  for (int m = 1; m <= l; ++m) {
    double sg = (m & 1) ? -1.0 : 1.0;
    q[(l + m) * d + (l + m)].re = sg * rs;
    q[(l + m) * d + (l - m)].im = sg * rs;
  }
  cplx ph;
  switch (l & 3) {                          // (-i)^l
    case 0: ph.re = 1;  ph.im = 0;  break;
    case 1: ph.re = 0;  ph.im = -1; break;
    case 2: ph.re = -1; ph.im = 0;  break;
    default: ph.re = 0; ph.im = 1;  break;
  }
  for (int i = 0; i < d * d; ++i) q[i] = cmul(q[i], ph);
}

// combo tables: (l, l1, l2), sizes S = K*P*Q, prefix offsets into ws
// combos: (0,0,0)(0,2,2)(0,4,4)(2,0,2)(2,2,2)(2,2,4)(2,4,4)(4,0,4)(4,2,2)(4,2,4)(4,4,4)
#define CG_TOTAL 2327

__global__ void cg_setup_kernel(float* __restrict__ ws) {
  __shared__ double Cc[729];
  __shared__ cplx Q1s[81], Q2s[81], Q3s[81];
  __shared__ double Rre[729], Rim[729];
  __shared__ double s_scale;
  __shared__ int s_useIm;

  const int Ls[11]  = {0, 0, 0, 2, 2, 2, 2, 4, 4, 4, 4};
  const int L1s[11] = {0, 2, 4, 0, 2, 2, 4, 0, 2, 2, 4};
  const int L2s[11] = {0, 2, 4, 2, 2, 4, 4, 4, 2, 4, 4};
  const int OFF[11] = {0, 1, 26, 107, 132, 257, 482, 887, 968, 1193, 1598};

  int combo = blockIdx.x;
  int l = Ls[combo], l1 = L1s[combo], l2 = L2s[combo];
  int K = 2 * l + 1, P = 2 * l1 + 1, Q = 2 * l2 + 1;
  int tid = threadIdx.x, nt = blockDim.x;
  int PQK = P * Q * K;

  // Cc[i][k][m] = su2_cg(l1, i-l1, l2, k-l2, l, m-l)
  for (int t = tid; t < PQK; t += nt) {
    int i = t / (Q * K); int rr = t % (Q * K); int k = rr / K; int m = rr % K;
    Cc[t] = su2_cg(l1, i - l1, l2, k - l2, l, m - l);
  }
  if (tid == 0) { buildQ(l1, Q1s); buildQ(l2, Q2s); buildQ(l, Q3s); }
  __syncthreads();

  // R[j,l',n] = sum_{i,k,m} Q1[i][j] Q2[k][l'] conj(Q3[m][n]) Cc[i][k][m]
  for (int t = tid; t < PQK; t += nt) {
    int j = t / (Q * K); int rr = t % (Q * K); int lp = rr / K; int n = rr % K;
    double sre = 0.0, sim = 0.0;
    for (int i = 0; i < P; ++i) {
      cplx q1 = Q1s[i * P + j];
      for (int k = 0; k < Q; ++k) {
        cplx a = cmul(q1, Q2s[k * Q + lp]);
        for (int m = 0; m < K; ++m) {
          double c = Cc[(i * Q + k) * K + m];
          if (c == 0.0) continue;
          cplx q3 = Q3s[m * K + n]; q3.im = -q3.im;  // conj
          cplx b = cmul(a, q3);
          sre += b.re * c; sim += b.im * c;
        }
      }
    }
    Rre[t] = sre; Rim[t] = sim;
  }
  __syncthreads();
  if (tid == 0) {
    double nr = 0.0, ni = 0.0;
    for (int t = 0; t < PQK; ++t) { nr += Rre[t] * Rre[t]; ni += Rim[t] * Rim[t]; }
    int useIm = (sqrt(nr) < 1e-8) ? 1 : 0;
    s_useIm = useIm;
    s_scale = 1.0 / sqrt(useIm ? ni : nr);
  }
  __syncthreads();
  double scale = s_scale; int useIm = s_useIm;

  // w[p,q,k] flat (p*Q+q)*K+k ; cg_r flat[k*PQ+f] = w[f%P, f/P, k] (reshape scramble)
  // cg_l flat[k*PQ + (f%Q)*P + (f/Q)] = cg_r flat[k*PQ+f]
  float* cgr = ws + OFF[combo];
  float* cgl = ws + CG_TOTAL + OFF[combo];
  int PQ = P * Q;
  for (int t = tid; t < K * PQ; t += nt) {
    int k = t / PQ, f = t % PQ;
    int wp = f % P, wq = f / P;
    double src = useIm ? Rim[(wp * Q + wq) * K + k] : Rre[(wp * Q + wq) * K + k];
    float v = (float)(src * scale);
    cgr[k * PQ + f] = v;
    int p = f / Q, q = f % Q;
    cgl[k * PQ + q * P + p] = v;
  }
}

// ---------------- main kernel: 1 wave per channel, WMMA f32 16x16x4 z-GEMMs ----------------

__device__ inline float wave_sum32(float v) {
  #pragma unroll
  for (int o = 16; o; o >>= 1) v += __shfl_xor(v, o, 32);
  return v;
}

#define WAVE_LDS 4560  // floats per wave region

__global__ __launch_bounds__(64) void quad_main_kernel(
    const float* __restrict__ rh0, const float* __restrict__ rh2,
    const float* __restrict__ rh4, const float* __restrict__ cg,
    float* __restrict__ out0, float* __restrict__ out2, float* __restrict__ out4,
    float* __restrict__ ofeats, int M) {
  __shared__ float sm[2 * WAVE_LDS];
  int wave = threadIdx.x >> 5, lane = threadIdx.x & 31;
  int ch = blockIdx.x * 2 + wave;
  if (ch >= M) ch = M - 1;  // duplicate work (same values) instead of diverging past barriers

  float* wb  = sm + wave * WAVE_LDS;
  float* R0s = wb;            // 1
  float* R2s = wb + 1;        // 25
  float* R4s = wb + 26;       // 81
  float* Y4  = wb + 108;      // 9 x 160
  float* X4  = wb + 1548;     // 9 x 160
  float* Y2  = wb + 2988;     // 5 x 156
  float* X2  = wb + 3768;     // 5 x 156

  // stage channel data into LDS
  for (int t = lane; t < 81; t += 32) R4s[t] = rh4[(size_t)ch * 81 + t];
  if (lane < 25) R2s[lane] = rh2[(size_t)ch * 25 + lane];
  if (lane == 0) R0s[0] = rh0[ch];
  __syncthreads();

  const float* srcs[3] = {R0s, R2s, R4s};

  // ---- build Y4/X4 (K=9, width 160): combos (4,0,4)(4,2,2)(4,2,4)(4,4,4) ----
  {
    const int Pm[4] = {1, 5, 5, 9}, Qm[4] = {9, 5, 9, 9};
    const int Bm[4] = {0, 9, 34, 79}, Om[4] = {887, 968, 1193, 1598};
    const int S1[4] = {0, 1, 1, 2}, S2[4] = {2, 1, 2, 2};
    for (int c = 0; c < 4; ++c) {
      int P = Pm[c], Q = Qm[c], PQ = P * Q, base = Bm[c];
      const float* cgr = cg + Om[c];
      const float* cgl = cg + CG_TOTAL + Om[c];
      const float* RH1 = srcs[S1[c]];
      const float* RH2 = srcs[S2[c]];
      for (int t = lane; t < 9 * PQ; t += 32) {
        int k = t / PQ, r = t - k * PQ, p = r / Q, q = r - p * Q;
        const float* cr = cgr + (k * P + p) * Q;
        float sx = 0.f;
        for (int m = 0; m < Q; ++m) sx += RH2[q * Q + m] * cr[m];
        X4[k * 160 + base + r] = sx;
        const float* cl = cgl + (k * Q + q) * P;
        float sy = 0.f;
        for (int m = 0; m < P; ++m) sy += RH1[m * P + p] * cl[m];
        Y4[k * 160 + base + r] = sy;
      }
    }
  }
  // ---- build Y2/X2 (K=5, width 156): combos (2,0,2)(2,2,2)(2,2,4)(2,4,4) ----
  {
    const int Pm[4] = {1, 5, 5, 9}, Qm[4] = {5, 5, 9, 9};
    const int Bm[4] = {0, 5, 30, 75}, Om[4] = {107, 132, 257, 482};
    const int S1[4] = {0, 1, 1, 2}, S2[4] = {1, 1, 2, 2};
    for (int c = 0; c < 4; ++c) {
      int P = Pm[c], Q = Qm[c], PQ = P * Q, base = Bm[c];
      const float* cgr = cg + Om[c];
      const float* cgl = cg + CG_TOTAL + Om[c];
      const float* RH1 = srcs[S1[c]];
      const float* RH2 = srcs[S2[c]];
      for (int t = lane; t < 5 * PQ; t += 32) {
        int k = t / PQ, r = t - k * PQ, p = r / Q, q = r - p * Q;
        const float* cr = cgr + (k * P + p) * Q;
        float sx = 0.f;
        for (int m = 0; m < Q; ++m) sx += RH2[q * Q + m] * cr[m];
        X2[k * 156 + base + r] = sx;
        const float* cl = cgl + (k * Q + q) * P;
        float sy = 0.f;
        for (int m = 0; m < P; ++m) sy += RH1[m * P + p] * cl[m];
        Y2[k * 156 + base + r] = sy;
      }
    }
  }
  // ---- l=0 (K=1) directly with VALU: combos (0,0,0)(0,2,2)(0,4,4) ----
  float z0p = 0.f;
  {
    const int Pm[3] = {1, 5, 9}, Om[3] = {0, 1, 26}, Sc[3] = {0, 1, 2};
    for (int c = 0; c < 3; ++c) {
      int P = Pm[c], Q = P;
      const float* cgr = cg + Om[c];
      const float* cgl = cg + CG_TOTAL + Om[c];
      const float* RH = srcs[Sc[c]];
      for (int t = lane; t < P * Q; t += 32) {
        int p = t / Q, q = t - p * Q;
        float sx = 0.f;
        for (int m = 0; m < Q; ++m) sx += RH[q * Q + m] * cgr[p * Q + m];
        float sy = 0.f;
        for (int m = 0; m < P; ++m) sy += RH[m * P + p] * cgl[q * P + m];
        z0p += sx * sy;
      }
    }
  }
  float z0 = wave_sum32(z0p);
  __syncthreads();

  // ---- WMMA chains: Z = Y * X^T, fp32 16x16x4, accumulate along concatenated PQ axis ----
  int row = lane & 15, hf = lane >> 4;
  v8f acc4 = {0, 0, 0, 0, 0, 0, 0, 0};
  for (int t = 0; t < 40; ++t) {
    int col = 4 * t + 2 * hf;
    v2f a = *(const v2f*)(Y4 + row * 160 + col);   // A: Y rows (M=k), K-dim cols
    v2f b = *(const v2f*)(X4 + row * 160 + col);   // B: X^T -> lane = N col of X row
    acc4 = __builtin_amdgcn_wmma_f32_16x16x4_f32(false, a, false, b, (short)0, acc4, false, false);
  }
  v8f acc2 = {0, 0, 0, 0, 0, 0, 0, 0};
  for (int t = 0; t < 39; ++t) {
    int col = 4 * t + 2 * hf;
    v2f a = *(const v2f*)(Y2 + row * 156 + col);
    v2f b = *(const v2f*)(X2 + row * 156 + col);
    acc2 = __builtin_amdgcn_wmma_f32_16x16x4_f32(false, a, false, b, (short)0, acc2, false, false);
  }

  // ---- store rh_n[4] (9x9) and sum of squares; D layout: (M = v + 8*hf, N = row) ----
  float ss4 = 0.f;
  if (row < 9) {
    float* o4 = out4 + (size_t)ch * 81;
    #pragma unroll
    for (int v = 0; v < 8; ++v) {
      int Mi = v + 8 * hf;
      if (Mi < 9) { float val = acc4[v]; o4[Mi * 9 + row] = val; ss4 += val * val; }
    }
  }
  ss4 = wave_sum32(ss4);
  float ss2 = 0.f;
  if (row < 5) {
    float* o2 = out2 + (size_t)ch * 25;
    #pragma unroll
    for (int v = 0; v < 8; ++v) {
      int Mi = v + 8 * hf;
      if (Mi < 5) { float val = acc2[v]; o2[Mi * 5 + row] = val; ss2 += val * val; }
    }
  }
  ss2 = wave_sum32(ss2);

  if (lane == 0) {
    out0[ch] = z0;
    int n = ch >> 3, abc = ch & 7;            // A*B*C = 8
    float* fr = ofeats + (size_t)n * 40;
    const float PI2_8 = 78.95683520871486f;   // 8*pi^2
    fr[16 + abc] = PI2_8 * z0 * z0;           // n_l = 8pi^2/(2l+1)
    fr[24 + abc] = (PI2_8 / 5.0f) * ss2;
    fr[32 + abc] = (PI2_8 / 9.0f) * ss4;
  }
}

// ---------------- feats passthrough ----------------

__global__ void feats_copy_kernel(const float* __restrict__ feats, float* __restrict__ ofeats, int total) {
  int i = blockIdx.x * blockDim.x + threadIdx.x;
  if (i < total) {
    int n = i >> 4, j = i & 15;
    ofeats[(size_t)n * 40 + j] = feats[i];
  }
}

// ---------------- launch ----------------

extern "C" void kernel_launch(void* const* d_in, const int* in_sizes, int n_in,
                              void* d_out, int out_size, void* d_ws, size_t ws_size,
                              hipStream_t stream) {
  const float* rh0   = (const float*)d_in[0];
  const float* rh2   = (const float*)d_in[1];
  const float* rh4   = (const float*)d_in[2];
  const float* feats = (const float*)d_in[3];
  float* out = (float*)d_out;
  float* ws  = (float*)d_ws;

  int M  = in_sizes[0];   // N*A*B*C = 65536 channels
  int NF = in_sizes[3];   // N*16 feats elements

  // d_out layout: [rh_n0 : M][rh_n2 : 25M][rh_n4 : 81M][out_feats : 40*(M/8) = 5M]
  float* out0   = out;
  float* out2   = out + (size_t)M;
  float* out4   = out + (size_t)26 * M;
  float* ofeats = out + (size_t)107 * M;

  cg_setup_kernel<<<11, 256, 0, stream>>>(ws);
  quad_main_kernel<<<(M + 1) / 2, 64, 0, stream>>>(rh0, rh2, rh4, ws, out0, out2, out4, ofeats, M);
  feats_copy_kernel<<<(NF + 255) / 256, 256, 0, stream>>>(feats, ofeats, NF);
}